// STNM_64604898066639
// MI455X (gfx1250) — compile-verified
//
#include <hip/hip_runtime.h>

// STNM: bilinear grid-sample + masked composite.  B=32, H=W=512, C=3, fp32 NHWC.
// Memory-bound (<0.4 FLOP/byte vs 23.3 TB/s HBM): optimize data movement.
//  - NT (non-temporal) b128 loads/stores for the streaming tensors so the
//    96 MB fgimg stays resident in the 192 MB L2 for the random gathers.
//  - RT (default) policy for fgimg gather loads -> ~4x L2 reuse.
//  - 4 pixels per lane => all streaming accesses are 16B-aligned b128.

typedef float v4f __attribute__((ext_vector_type(4)));

#define Bn   32
#define Hn   512
#define Wn   512
#define Cn   3
#define HWn  (Hn * Wn)          // 262144
#define NPIX (Bn * HWn)         // 8388608

__device__ __forceinline__ void gather3(const float* __restrict__ base,
                                        int yi, int xi, float w, float acc[3]) {
  // Out-of-bounds corner contributes zero (clamped address stays in-bounds).
  const bool inb = ((unsigned)yi < (unsigned)Hn) & ((unsigned)xi < (unsigned)Wn);
  const int yc = min(max(yi, 0), Hn - 1);
  const int xc = min(max(xi, 0), Wn - 1);
  const float* __restrict__ src = base + ((size_t)yc * Wn + xc) * Cn;
  const float wm = inb ? w : 0.0f;
  acc[0] = fmaf(wm, src[0], acc[0]);
  acc[1] = fmaf(wm, src[1], acc[1]);
  acc[2] = fmaf(wm, src[2], acc[2]);
}

__global__ __launch_bounds__(256) void stnm4_kernel(
    const float* __restrict__ canvas,
    const float* __restrict__ fgimg,
    const float* __restrict__ fggrid,
    const float* __restrict__ fgmask,
    float* __restrict__ out) {
  const long long t  = (long long)blockIdx.x * blockDim.x + threadIdx.x;
  const long long p0 = t * 4;  // first of 4 consecutive pixels
  if (p0 >= NPIX) return;

  // ---- streaming inputs: non-temporal b128 loads (don't pollute L2) ----
  const v4f g01 = __builtin_nontemporal_load((const v4f*)(fggrid + p0 * 2));     // y0 x0 y1 x1
  const v4f g23 = __builtin_nontemporal_load((const v4f*)(fggrid + p0 * 2 + 4)); // y2 x2 y3 x3
  const v4f mk  = __builtin_nontemporal_load((const v4f*)(fgmask + p0));
  const v4f c0  = __builtin_nontemporal_load((const v4f*)(canvas + p0 * 3));
  const v4f c1  = __builtin_nontemporal_load((const v4f*)(canvas + p0 * 3 + 4));
  const v4f c2  = __builtin_nontemporal_load((const v4f*)(canvas + p0 * 3 + 8));

  // All 4 pixels share one batch image (HW divisible by 4); HW is 2^18 -> shift.
  const int b = (int)(p0 / HWn);
  const float* __restrict__ fgb = fgimg + (size_t)b * HWn * Cn;

  const float gy[4] = {g01.x, g01.z, g23.x, g23.z};
  const float gx[4] = {g01.y, g01.w, g23.y, g23.w};
  const float mv[4] = {mk.x, mk.y, mk.z, mk.w};
  const float cv[12] = {c0.x, c0.y, c0.z, c0.w,
                        c1.x, c1.y, c1.z, c1.w,
                        c2.x, c2.y, c2.z, c2.w};
  float ov[12];

#pragma unroll
  for (int i = 0; i < 4; ++i) {
    const float yf = (gy[i] + 1.0f) * ((Hn - 1) * 0.5f);
    const float xf = (gx[i] + 1.0f) * ((Wn - 1) * 0.5f);
    const float y0 = floorf(yf);
    const float x0 = floorf(xf);
    const float wy = yf - y0;
    const float wx = xf - x0;
    const int y0i = (int)y0;
    const int x0i = (int)x0;

    float acc[3] = {0.0f, 0.0f, 0.0f};
    // fgimg gathers use default (RT) policy -> cached in L2 for reuse.
    gather3(fgb, y0i,     x0i,     (1.0f - wy) * (1.0f - wx), acc);
    gather3(fgb, y0i,     x0i + 1, (1.0f - wy) * wx,          acc);
    gather3(fgb, y0i + 1, x0i,     wy * (1.0f - wx),          acc);
    gather3(fgb, y0i + 1, x0i + 1, wy * wx,                   acc);

    const float m  = mv[i];
    const float om = 1.0f - m;
#pragma unroll
    for (int ch = 0; ch < 3; ++ch)
      ov[3 * i + ch] = m * acc[ch] + om * cv[3 * i + ch];
  }

  // ---- output: non-temporal b128 stores (write-once, never re-read) ----
  const v4f o0 = {ov[0], ov[1], ov[2],  ov[3]};
  const v4f o1 = {ov[4], ov[5], ov[6],  ov[7]};
  const v4f o2 = {ov[8], ov[9], ov[10], ov[11]};
  __builtin_nontemporal_store(o0, (v4f*)(out + p0 * 3));
  __builtin_nontemporal_store(o1, (v4f*)(out + p0 * 3 + 4));
  __builtin_nontemporal_store(o2, (v4f*)(out + p0 * 3 + 8));
}

extern "C" void kernel_launch(void* const* d_in, const int* in_sizes, int n_in,
                              void* d_out, int out_size, void* d_ws, size_t ws_size,
                              hipStream_t stream) {
  (void)in_sizes; (void)n_in; (void)out_size; (void)d_ws; (void)ws_size;
  const float* canvas = (const float*)d_in[0];
  const float* fgimg  = (const float*)d_in[1];
  const float* fggrid = (const float*)d_in[2];
  const float* fgmask = (const float*)d_in[3];
  float* out = (float*)d_out;

  const int threads = 256;                 // 8 wave32s per block
  const int work    = NPIX / 4;            // 4 pixels per lane, exact division
  const int blocks  = work / threads;      // 8192 blocks
  stnm4_kernel<<<blocks, threads, 0, stream>>>(canvas, fgimg, fggrid, fgmask, out);
}